// FullyConnectedTensorProduct_46497315947111
// MI455X (gfx1250) — compile-verified
//
#include <hip/hip_runtime.h>

typedef __attribute__((ext_vector_type(2))) float v2f;
typedef __attribute__((ext_vector_type(8))) float v8f;
typedef __attribute__((ext_vector_type(4))) unsigned int v4u;
typedef __attribute__((ext_vector_type(4))) int v4i;
typedef __attribute__((ext_vector_type(8))) int v8i;

#define BATCH 8192
#define XDIM  480

// ---------------- path tables (compile-time) ----------------
constexpr int NP = 11;
// path order: (0,0,0)(0,1,1)(0,2,2)(1,0,1)(1,1,0)(1,1,2)(1,2,1)(2,0,2)(2,1,1)(2,2,0)(2,2,2)
constexpr int PM1[NP]   = {128,128,128, 64, 64, 64, 64, 32, 32, 32, 32};
constexpr int PM2[NP]   = {128, 64, 32,128, 64, 64, 32,128, 64, 32, 32};
constexpr int PM3[NP]   = {128, 64, 32, 64,128, 32, 64, 32, 64,128, 32};
constexpr int PD1[NP]   = {1,1,1, 3,3,3,3, 5,5,5,5};
constexpr int PD2[NP]   = {1,3,5, 1,3,3,5, 1,3,5,5};
constexpr int PD3[NP]   = {1,3,5, 3,1,5,3, 5,3,1,5};
constexpr int POFF1[NP] = {0,0,0, 128,128,128,128, 320,320,320,320};
constexpr int POFF2[NP] = {0,128,320, 0,128,128,320, 0,128,320,320};
constexpr int POFF3[NP] = {0,128,320, 128,0,320,128, 320,128,0,320};
constexpr long long PWOFF[NP] = {0,2097152,2621440,2752512,3276800,3801088,
                                 3932160,4063232,4194304,4325376,4456448};
// coeff = sqrt((2*l3+1)/fan_in[i3]); fan_in = {21504, 20480, 13312}
constexpr float C_I0 = 0.00681934f, C_I1 = 0.01210307f, C_I2 = 0.01938040f;
constexpr float PCOEFF[NP] = {C_I0,C_I1,C_I2, C_I1,C_I0,C_I2,C_I1, C_I2,C_I1,C_I0,C_I2};

// ---------------- CG nonzeros (unit Frobenius norm, e3nn real basis) ----------------
constexpr float R3 = 0.57735027f;
constexpr float R5 = 0.44721360f;
constexpr float AL = 0.31622777f;   // 1/sqrt(10)
constexpr float B2 = 0.36514837f;   // 2/sqrt(30)
constexpr float B1 = 0.18257419f;   // 1/sqrt(30)
constexpr float A2 = 0.23904572f;   // sqrt(2/35)
constexpr float H2 = 0.11952286f;   // 0.5*sqrt(2/35)
constexpr float G2 = 0.20701967f;   // 0.5*sqrt(6/35)

constexpr int NZ_OFF[NP+1] = {0,1,4,9,12,15,26,37,42,53,58,83};
constexpr int NZI[83] = {
  0,
  0,0,0,
  0,0,0,0,0,
  0,1,2,
  0,1,2,
  2,0,0,1,1,0,2,1,2,2,0,
  2,0,0,1,1,0,2,1,2,2,0,
  0,1,2,3,4,
  0,0,1,1,2,2,2,3,3,4,4,
  0,1,2,3,4,
  0,0,2,4,4,2,2,1,1,2,3,3,2,1,1,4,3,3,4,0,0,1,1,3,3 };
constexpr int NZJ[83] = {
  0,
  0,1,2,
  0,1,2,3,4,
  0,0,0,
  0,1,2,
  0,2,1,0,1,0,2,2,1,2,0,
  0,0,1,1,2,2,2,3,3,4,4,
  0,0,0,0,0,
  2,0,0,1,1,0,2,1,2,2,0,
  0,1,2,3,4,
  0,2,0,4,2,4,2,1,2,1,3,2,3,1,4,1,3,4,3,1,3,0,3,0,1 };
constexpr int NZK[83] = {
  0,
  0,1,2,
  0,1,2,3,4,
  0,1,2,
  0,0,0,
  0,0,1,1,2,2,2,3,3,4,4,
  0,2,1,0,1,0,2,2,1,2,0,
  0,1,2,3,4,
  0,2,1,0,1,0,2,2,1,2,0,
  0,0,0,0,0,
  2,0,0,2,4,4,2,2,1,1,2,3,3,4,1,1,4,3,3,3,1,3,0,1,0 };
constexpr float NZC[83] = {
  1.0f,
  R3,R3,R3,
  R5,R5,R5,R5,R5,
  R3,R3,R3,
  R3,R3,R3,
  AL,AL,AL,AL,B2,-B1,-B1,AL,AL,AL,-AL,
  AL,AL,AL,AL,B2,-B1,-B1,AL,AL,AL,-AL,
  R5,R5,R5,R5,R5,
  AL,AL,AL,AL,B2,-B1,-B1,AL,AL,AL,-AL,
  R5,R5,R5,R5,R5,
  -A2,-A2,-A2,-A2,-A2,-A2,A2, H2,H2,H2, H2,H2,H2, -G2,-G2,-G2, G2,G2,G2, G2,G2,G2,G2,G2,G2 };

// ---------------- Tensor Data Mover: 2D tile load, f32 elements ----------------
__device__ __forceinline__ void tdm_load_2d(unsigned lds_addr, const void* gptr,
                                            unsigned tile_d0, unsigned tile_d1,
                                            unsigned tensor_d0, unsigned tensor_d1,
                                            unsigned long long stride0_elems)
{
  unsigned long long ga = (unsigned long long)(uintptr_t)gptr;
  v4u g0;
  g0[0] = 1u;                                        // count=1, user mode, no gather
  g0[1] = lds_addr;                                  // LDS byte address
  g0[2] = (unsigned)(ga & 0xffffffffu);              // global_addr[31:0]
  g0[3] = (unsigned)((ga >> 32) & 0x01ffffffu) | (2u << 30); // addr[56:32] | type=2
  v8i g1;
  g1[0] = (int)(2u << 16);                           // wg_mask=0, data_size=4B
  g1[1] = (int)((tensor_d0 & 0xffffu) << 16);        // tensor_dim0[15:0]
  g1[2] = (int)((tensor_d0 >> 16) | ((tensor_d1 & 0xffffu) << 16));
  g1[3] = (int)((tensor_d1 >> 16) | (tile_d0 << 16));// tensor_dim1[31:16] | tile_dim0
  g1[4] = (int)(tile_d1);                            // tile_dim1 | tile_dim2(0)<<16
  g1[5] = (int)(stride0_elems & 0xffffffffu);        // tensor_dim0_stride[31:0]
  g1[6] = (int)((stride0_elems >> 32) & 0xffffu);    // stride0[47:32] | stride1(0)
  g1[7] = 0;
  v4i z4 = {0, 0, 0, 0};
#if defined(__clang_major__) && (__clang_major__ >= 23)
  v8i z8 = {0, 0, 0, 0, 0, 0, 0, 0};
  __builtin_amdgcn_tensor_load_to_lds(g0, g1, z4, z4, z8, 0);
#else
  __builtin_amdgcn_tensor_load_to_lds(g0, g1, z4, z4, 0);
#endif
}

__device__ __forceinline__ unsigned lds_off(const void* p) {
  return (unsigned)(uintptr_t)p;   // LDS aperture: addr[31:0] is the LDS byte offset
}

// ---------------- kernels ----------------
__global__ __launch_bounds__(256) void zero_out_kernel(float* __restrict__ out, int n) {
  int i = blockIdx.x * 256 + threadIdx.x;
  if (i < n) out[i] = 0.0f;
}

template <int P>
__global__ __launch_bounds__(256)
void tp_path_kernel(const float* __restrict__ x1, const float* __restrict__ x2,
                    const float* __restrict__ w,  float* __restrict__ out)
{
  constexpr int m2  = PM2[P], m3 = PM3[P];
  constexpr int d1  = PD1[P], d2 = PD2[P], d3 = PD3[P];
  constexpr int sz1 = PM1[P] * d1, sz2 = m2 * d2;
  constexpr int KTOT = PM1[P] * m2;
  constexpr int NT  = m3 / 16;                 // one N-tile (16 cols) per wave
  constexpr int KB  = 64;                      // K-block (uv rows per W stage)
  constexpr int NKB = KTOT / KB;
  constexpr int KBW = KB * 16;                 // floats per W slice buffer
  constexpr int UPB = (KB >= m2) ? (KB / m2) : 1;   // u-rows per block
  constexpr int VS  = ((KB < m2) ? KB : m2) / 4;    // v-steps per u-chunk
  constexpr int nz0 = NZ_OFF[P];
  constexpr int nzN = NZ_OFF[P + 1] - NZ_OFF[P];

  __shared__ float s1[16 * sz1];
  __shared__ float s2[16 * sz2];
  __shared__ float sw[NT * 2 * KBW];           // per-wave double-buffered W slices

  const int lane  = threadIdx.x;
  const int half  = lane >> 4;
  const int l16   = lane & 15;
  const int ty    = threadIdx.y;
  const int mrow0 = blockIdx.x * 16;
  const int ncol0 = ty * 16;

  // --- async stage-in: x tiles (wave 0) + first W slice (every wave) via TDM ---
  if (ty == 0) {
    tdm_load_2d(lds_off(&s1[0]), x1 + (size_t)mrow0 * XDIM + POFF1[P],
                sz1, 16, sz1, 16, XDIM);
    tdm_load_2d(lds_off(&s2[0]), x2 + (size_t)mrow0 * XDIM + POFF2[P],
                sz2, 16, sz2, 16, XDIM);
  }
  const float* wpath = w + PWOFF[P] + ncol0;   // this wave's 16 output columns
  float* swb = &sw[ty * 2 * KBW];
  tdm_load_2d(lds_off(swb), wpath, 16, KB, m3, KTOT, m3);

  v8f acc[d3];
  v8f vzero = {};
#pragma unroll
  for (int k = 0; k < d3; ++k) acc[k] = vzero;

  for (int kb = 0; kb < NKB; ++kb) {
    // prefetch next W slice into the other buffer, then wait for current
    if (kb + 1 < NKB) {
      tdm_load_2d(lds_off(swb + ((kb + 1) & 1) * KBW),
                  wpath + (size_t)(kb + 1) * KB * m3, 16, KB, m3, KTOT, m3);
      __builtin_amdgcn_s_wait_tensorcnt((short)1);   // current slice (and x) done
    } else {
      __builtin_amdgcn_s_wait_tensorcnt((short)0);
    }
    asm volatile("" ::: "memory");
    if (kb == 0) __syncthreads();                    // x tiles visible to all waves

    const float* wbuf = swb + (kb & 1) * KBW;
    const float* s1r  = &s1[l16 * sz1];
    const float* s2r  = &s2[l16 * sz2];

#pragma unroll
    for (int ur = 0; ur < UPB; ++ur) {
      const int u     = (KB >= m2) ? (kb * UPB + ur) : ((kb * KB) / m2);
      const int vbase = (KB >= m2) ? 0 : ((kb * KB) % m2);

      float x1v[d1];
#pragma unroll
      for (int i = 0; i < d1; ++i) x1v[i] = s1r[u * d1 + i];

#pragma unroll 4
      for (int vt = 0; vt < VS; ++vt) {
        const int v0 = vbase + vt * 4 + 2 * half;
        // x2 rows v0 and v0+1: 2*d2 contiguous floats
        float x2v[2 * d2];
#pragma unroll
        for (int j = 0; j < 2 * d2; ++j) x2v[j] = s2r[v0 * d2 + j];

        float av[d3][2];
#pragma unroll
        for (int k = 0; k < d3; ++k) { av[k][0] = 0.f; av[k][1] = 0.f; }
#pragma unroll
        for (int e = 0; e < 2; ++e) {
#pragma unroll
          for (int t = 0; t < nzN; ++t) {
            av[NZK[nz0 + t]][e] += NZC[nz0 + t] * x1v[NZI[nz0 + t]] * x2v[e * d2 + NZJ[nz0 + t]];
          }
        }

        const int kr = (ur * VS + vt) * 4 + 2 * half;   // local K row in W slice
        v2f b;
        b.x = wbuf[kr * 16 + l16];
        b.y = wbuf[(kr + 1) * 16 + l16];

#pragma unroll
        for (int k = 0; k < d3; ++k) {
          v2f a;
          a.x = av[k][0];
          a.y = av[k][1];
          acc[k] = __builtin_amdgcn_wmma_f32_16x16x4_f32(
              false, a, false, b, (short)0, acc[k], false, false);
        }
      }
    }
  }

  // epilogue: C layout vgpr r -> M = r + 8*half, lane -> N = l16
#pragma unroll
  for (int k = 0; k < d3; ++k) {
#pragma unroll
    for (int r = 0; r < 8; ++r) {
      size_t o = (size_t)(mrow0 + r + 8 * half) * XDIM
               + POFF3[P] + (size_t)(ncol0 + l16) * d3 + k;
      out[o] += PCOEFF[P] * acc[k][r];
    }
  }
}

// ---------------- launch ----------------
extern "C" void kernel_launch(void* const* d_in, const int* in_sizes, int n_in,
                              void* d_out, int out_size, void* d_ws, size_t ws_size,
                              hipStream_t stream) {
  (void)in_sizes; (void)n_in; (void)d_ws; (void)ws_size; (void)out_size;
  const float* x1 = (const float*)d_in[0];
  const float* x2 = (const float*)d_in[1];
  const float* w  = (const float*)d_in[2];
  float* out = (float*)d_out;

  const int n = BATCH * XDIM;
  zero_out_kernel<<<(n + 255) / 256, 256, 0, stream>>>(out, n);

#define LAUNCH_PATH(P)                                                         \
  tp_path_kernel<P><<<dim3(BATCH / 16, 1), dim3(32, PM3[P] / 16), 0, stream>>>(\
      x1, x2, w, out)

  LAUNCH_PATH(0);
  LAUNCH_PATH(1);
  LAUNCH_PATH(2);
  LAUNCH_PATH(3);
  LAUNCH_PATH(4);
  LAUNCH_PATH(5);
  LAUNCH_PATH(6);
  LAUNCH_PATH(7);
  LAUNCH_PATH(8);
  LAUNCH_PATH(9);
  LAUNCH_PATH(10);
#undef LAUNCH_PATH
}